// ErrorLSTMCell_66554813218870
// MI455X (gfx1250) — compile-verified
//
#include <hip/hip_runtime.h>
#include <math.h>

#define DD 4096
#define TWO_D 8192
#define K_PER_WAVE 1024   // 8 waves * 1024 = 8192 = 2*D

typedef float v2f __attribute__((ext_vector_type(2)));
typedef float v8f __attribute__((ext_vector_type(8)));

// -----------------------------------------------------------------------------
// Stage 1: gate pre-activations  g[gate][n] = sum_k a[k]*W[k][n] + b[n]
//   a = concat(h0, x)  (length 2D), W row-major [2D, D].
// One block = one (gate, 16-column tile). 8 waves split K, WMMA f32 16x16x4
// accumulates; A-operand rows are all identical (only row 0 consumed), B is the
// 4x16 weight tile streamed non-temporally from HBM.
// -----------------------------------------------------------------------------
__global__ __launch_bounds__(256) void lstm_gemv_wmma(
    const float* __restrict__ x,  const float* __restrict__ h0,
    const float* __restrict__ Wf, const float* __restrict__ bf,
    const float* __restrict__ Wi, const float* __restrict__ bi,
    const float* __restrict__ Wc, const float* __restrict__ bc,
    const float* __restrict__ Wo, const float* __restrict__ bo,
    float* __restrict__ gates /* [4][D] pre-activations */)
{
    const int gate = blockIdx.y;
    const float* W; const float* b;
    if      (gate == 0) { W = Wf; b = bf; }
    else if (gate == 1) { W = Wi; b = bi; }
    else if (gate == 2) { W = Wc; b = bc; }
    else                { W = Wo; b = bo; }

    const int n0   = blockIdx.x * 16;
    const int lane = threadIdx.x & 31;
    const int wave = threadIdx.x >> 5;

    // Each wave's K range lies entirely in h0 (waves 0..3) or x (waves 4..7).
    const int kstart = wave * K_PER_WAVE;
    const float* aseg = (kstart < DD) ? (h0 + kstart) : (x + (kstart - DD));

    // Assumed f32 operand layout (mirrors documented 16x4 A layout):
    //   VGPR0 holds K = 0 (lanes 0-15) / K = 2 (lanes 16-31)
    //   VGPR1 holds K = 1 (lanes 0-15) / K = 3 (lanes 16-31)
    //   B: N = lane & 15
    const int krow = (lane >> 4) << 1;   // 0 or 2
    const float* wp = W + (size_t)(kstart + krow) * DD + n0 + (lane & 15);

    v8f acc = {};

    #pragma unroll 4
    for (int o = 0; o < K_PER_WAVE; o += 4) {
        // A operand: every lane carries the same a-chunk pair (rows 1..15 of
        // the 16x4 A tile duplicate row 0 -> duplicate garbage rows in D that
        // we never read; bounded values, no NaN risk). Small cached load.
        v2f av = *(const v2f*)(aseg + o + krow);

        // B operand: two coalesced non-temporal b32 loads (rows k, k+1 of the
        // 4x16 W tile for this lane's K pair). Streamed once -> NT hint keeps
        // the 512 MB weight stream from thrashing L2.
        v2f bv;
        bv.x = __builtin_nontemporal_load(wp);
        bv.y = __builtin_nontemporal_load(wp + DD);
        wp += (size_t)4 * DD;

        acc = __builtin_amdgcn_wmma_f32_16x16x4_f32(
            false, av, false, bv, (short)0, acc, false, false);
    }

    // Row M=0 of D lives in acc element 0, lanes 0..15 (N = lane).
    __shared__ float red[8 * 16];
    if (lane < 16) red[wave * 16 + lane] = acc[0];
    __syncthreads();

    if (threadIdx.x < 16) {
        float s = b[n0 + threadIdx.x];
        #pragma unroll
        for (int w = 0; w < 8; ++w) s += red[w * 16 + threadIdx.x];
        gates[(size_t)gate * DD + n0 + threadIdx.x] = s;
    }
}

// -----------------------------------------------------------------------------
// Stage 2: pointwise LSTM gate math. 4096 elements, trivially bandwidth-free.
// -----------------------------------------------------------------------------
__global__ __launch_bounds__(256) void lstm_pointwise(
    const float* __restrict__ gates, const float* __restrict__ c0,
    float* __restrict__ h)
{
    int n = blockIdx.x * blockDim.x + threadIdx.x;
    if (n >= DD) return;
    float gf = gates[0 * DD + n];
    float gi = gates[1 * DD + n];
    float gc = gates[2 * DD + n];
    float go = gates[3 * DD + n];

    float zf = 1.0f / (1.0f + expf(-gf));
    float zi = 1.0f / (1.0f + expf(-gi));
    float zo = 1.0f / (1.0f + expf(-go));
    float z  = tanhf(gc);

    float c = c0[n] * zf + z * zi;
    h[n] = zo * tanhf(c);
}

extern "C" void kernel_launch(void* const* d_in, const int* in_sizes, int n_in,
                              void* d_out, int out_size, void* d_ws, size_t ws_size,
                              hipStream_t stream) {
    // setup_inputs order: x, h0, c0, Wf, bf, Wi, bi, Wc, bc, Wo, bo (all f32)
    const float* x  = (const float*)d_in[0];
    const float* h0 = (const float*)d_in[1];
    const float* c0 = (const float*)d_in[2];
    const float* Wf = (const float*)d_in[3];
    const float* bf = (const float*)d_in[4];
    const float* Wi = (const float*)d_in[5];
    const float* bi = (const float*)d_in[6];
    const float* Wc = (const float*)d_in[7];
    const float* bc = (const float*)d_in[8];
    const float* Wo = (const float*)d_in[9];
    const float* bo = (const float*)d_in[10];

    float* gates = (float*)d_ws;     // 4 * 4096 floats = 64 KiB of scratch
    float* h     = (float*)d_out;    // 4096 floats

    dim3 grid1(DD / 16, 4);          // 256 column tiles x 4 gates = 1024 blocks
    lstm_gemv_wmma<<<grid1, 256, 0, stream>>>(x, h0, Wf, bf, Wi, bi,
                                              Wc, bc, Wo, bo, gates);

    lstm_pointwise<<<(DD + 255) / 256, 256, 0, stream>>>(gates, c0, h);
}